// Net_41008347742650
// MI455X (gfx1250) — compile-verified
//
#include <hip/hip_runtime.h>
#include <math.h>

typedef float v2f __attribute__((ext_vector_type(2)));
typedef float v8f __attribute__((ext_vector_type(8)));

// ---------------------------------------------------------------------------
// Image gradients -> per-pixel pose (unit gradient dir) and act (|grad|)
// jnp.gradient semantics: central diff interior, one-sided at borders.
// ---------------------------------------------------------------------------
__global__ void grad_kernel(const float* __restrict__ img, float2* __restrict__ pose,
                            float* __restrict__ act, int total) {
  for (int i = blockIdx.x * blockDim.x + threadIdx.x; i < total;
       i += gridDim.x * blockDim.x) {
    int x = i & 31, y = (i >> 5) & 31, b = i >> 10;
    const float* g = img + b * 1024;
    float gx, gy;
    if (x == 0)       gx = g[y*32 + 1]  - g[y*32 + 0];
    else if (x == 31) gx = g[y*32 + 31] - g[y*32 + 30];
    else              gx = 0.5f * (g[y*32 + x + 1] - g[y*32 + x - 1]);
    if (y == 0)       gy = g[32 + x]    - g[x];
    else if (y == 31) gy = g[31*32 + x] - g[30*32 + x];
    else              gy = 0.5f * (g[(y+1)*32 + x] - g[(y-1)*32 + x]);
    float mag = sqrtf(gx*gx + gy*gy + 1e-12f);
    pose[i] = make_float2(gx / mag, gy / mag);
    act[i]  = mag;
  }
}

// ---------------------------------------------------------------------------
// f32 WMMA GEMM: C[M x NC] = A[M x Kd] * B[Kd x NC], all row-major.
// LAUNCH REQUIREMENTS (all satisfied by kernel_launch below):
//   M % 64 == 0   (4 waves per block, one 16x16 tile each along M)
//   Kd % 16 == 0  (K-loop unrolled 4x, step 4)
//   NC ragged OK  (B column clamped; edge lanes masked at store)
// Inner loop is branchless, full EXEC -> clean v_wmma_f32_16x16x4_f32 stream.
// ---------------------------------------------------------------------------
__global__ void __launch_bounds__(128) wmma_gemm_f32(
    const float* __restrict__ A, const float* __restrict__ B,
    float* __restrict__ C, int M, int Kd, int NC) {
  const int lane = threadIdx.x & 31;
  const int wave = threadIdx.x >> 5;
  const int half = lane >> 4;          // 0 or 1
  const int l16  = lane & 15;
  const int m0 = (blockIdx.y * 4 + wave) * 16;
  const int n0 = blockIdx.x * 16;

  // Clamp B column for ragged NC; clamped lanes discarded at store.
  const int bc = min(n0 + l16, NC - 1);
  const float* __restrict__ Arow = A + (m0 + l16) * Kd + half * 2;  // 8B aligned
  const float* __restrict__ Bcol = B + bc;

  v8f acc = {};
#pragma unroll 4
  for (int k = 0; k < Kd; k += 4) {
    // A frag (ISA 7.12.2): VGPR0=K(k+2*half), VGPR1=K(k+2*half+1), row m0+l16
    v2f a = *(const v2f*)(Arow + k);                 // global_load_b64
    // B frag: rows k+2*half, k+2*half+1; col n0+l16
    const float* bp = Bcol + (k + half * 2) * NC;
    v2f b;
    b.x = bp[0];
    b.y = bp[NC];
    acc = __builtin_amdgcn_wmma_f32_16x16x4_f32(
        /*neg_a=*/false, a, /*neg_b=*/false, b,
        /*c_mod=*/(short)0, acc, /*reuse_a=*/false, /*reuse_b=*/false);
  }

  const int cn = n0 + l16;
  if (cn < NC) {
#pragma unroll
    for (int r = 0; r < 8; ++r) {
      int cm = m0 + r + half * 8;      // C rows r / r+8 across lane halves
      C[cm * NC + cn] = acc[r];
    }
  }
}

// ---------------------------------------------------------------------------
// Permute conv weight (5,5,ci,co) -> Bperm[ci][25*co] so gconv GEMM is
// Y[n, t*co+o] = sum_c X[n,c] * W[t,c,o].
// ---------------------------------------------------------------------------
__global__ void permW(const float* __restrict__ W, float* __restrict__ Bm,
                      int ci, int co, int total) {
  for (int i = blockIdx.x * blockDim.x + threadIdx.x; i < total;
       i += gridDim.x * blockDim.x) {
    int o  = i % co;
    int t2 = i / co;
    int c  = t2 % ci;
    int t  = t2 / ci;
    Bm[c * 25 * co + t * co + o] = W[i];
  }
}

// bilinear tap weights from pose-rotated offset (K=5 grid)
__device__ __forceinline__ void taps(float c, float s, int dx, int dy,
                                     int& t00, int& t01, int& t10, int& t11,
                                     float& w00, float& w01, float& w10, float& w11) {
  float fdx = (float)dx, fdy = (float)dy;
  float rx = c * fdx + s * fdy;
  float ry = -s * fdx + c * fdy;
  float uy = fminf(fmaxf((ry + 1.0f) * 2.0f, 0.0f), 4.0f);
  float ux = fminf(fmaxf((rx + 1.0f) * 2.0f, 0.0f), 4.0f);
  float y0f = floorf(uy), x0f = floorf(ux);
  float fy = uy - y0f, fx = ux - x0f;
  int y0 = (int)y0f, x0 = (int)x0f;
  int y1 = min(y0 + 1, 4), x1 = min(x0 + 1, 4);
  t00 = y0 * 5 + x0; t01 = y0 * 5 + x1;
  t10 = y1 * 5 + x0; t11 = y1 * 5 + x1;
  w00 = (1.0f - fy) * (1.0f - fx); w01 = (1.0f - fy) * fx;
  w10 = fy * (1.0f - fx);          w11 = fy * fx;
}

// ---------------------------------------------------------------------------
// Level-1 gconv (ci=1, co=16, 32x32 grid): direct, no GEMM needed.
// ---------------------------------------------------------------------------
__global__ void gconv_l1(const float* __restrict__ img, const float2* __restrict__ pose,
                         const float* __restrict__ Wc, float* __restrict__ out, int total) {
  for (int i = blockIdx.x * blockDim.x + threadIdx.x; i < total;
       i += gridDim.x * blockDim.x) {
    int o = i & 15;
    int node = i >> 4;
    int x = node & 31, y = (node >> 5) & 31, b = node >> 10;
    float2 ps = pose[node];
    float c = ps.x, s = ps.y;
    float acc = 0.0f;
    for (int dy = -1; dy <= 1; ++dy)
      for (int dx = -1; dx <= 1; ++dx) {
        int yy = y + dy, xx = x + dx;
        if ((unsigned)yy >= 32u || (unsigned)xx >= 32u) continue;
        float gv = img[b * 1024 + yy * 32 + xx];
        int t00, t01, t10, t11; float w00, w01, w10, w11;
        taps(c, s, dx, dy, t00, t01, t10, t11, w00, w01, w10, w11);
        acc += gv * (w00 * Wc[t00 * 16 + o] + w01 * Wc[t01 * 16 + o] +
                     w10 * Wc[t10 * 16 + o] + w11 * Wc[t11 * 16 + o]);
      }
    out[i] = fmaxf(acc, 0.0f);
  }
}

// ---------------------------------------------------------------------------
// gconv aggregation (levels 2-5): gather 4 precomputed tap rows of Y per edge.
// out[row,o] = relu( sum_{nbr} sum_{tap} w_tap * Y[col, tap*co + o] )
// ---------------------------------------------------------------------------
__global__ void gconv_agg(const float* __restrict__ Y, const float2* __restrict__ pose,
                          float* __restrict__ out, int H, int co, int total) {
  for (int i = blockIdx.x * blockDim.x + threadIdx.x; i < total;
       i += gridDim.x * blockDim.x) {
    int o = i % co;
    int node = i / co;
    int x = node % H;
    int y = (node / H) % H;
    int b = node / (H * H);
    float2 ps = pose[node];
    float c = ps.x, s = ps.y;
    float acc = 0.0f;
    int stride = 25 * co;
    for (int dy = -1; dy <= 1; ++dy)
      for (int dx = -1; dx <= 1; ++dx) {
        int yy = y + dy, xx = x + dx;
        if ((unsigned)yy >= (unsigned)H || (unsigned)xx >= (unsigned)H) continue;
        int col = (b * H + yy) * H + xx;
        int t00, t01, t10, t11; float w00, w01, w10, w11;
        taps(c, s, dx, dy, t00, t01, t10, t11, w00, w01, w10, w11);
        const float* Yc = Y + col * stride + o;
        acc += w00 * Yc[t00 * co] + w01 * Yc[t01 * co] +
               w10 * Yc[t10 * co] + w11 * Yc[t11 * co];
      }
    out[i] = fmaxf(acc, 0.0f);
  }
}

// ---------------------------------------------------------------------------
// 2x2 capsule pool with 1 routing iter. XT = xconv @ Wf already computed.
// Thread per (parent, o); o==0 lane also updates pose/act.
// ---------------------------------------------------------------------------
__global__ void pool_kernel(const float* __restrict__ XT, const float2* __restrict__ pose_in,
                            const float* __restrict__ act_in, float* __restrict__ x_out,
                            float2* __restrict__ pose_out, float* __restrict__ act_out,
                            int H, int fo, int total) {
  int H2 = H >> 1;
  for (int i = blockIdx.x * blockDim.x + threadIdx.x; i < total;
       i += gridDim.x * blockDim.x) {
    int o = i % fo;
    int p = i / fo;
    int X = p % H2;
    int Yp = (p / H2) % H2;
    int b = p / (H2 * H2);
    int base = b * H * H;
    int ch[4];
    float a[4];
    float amax = -1e30f;
    for (int cidx = 0; cidx < 4; ++cidx) {
      int dy = cidx >> 1, dx = cidx & 1;
      ch[cidx] = base + (2 * Yp + dy) * H + (2 * X + dx);
      a[cidx] = fmaxf(act_in[ch[cidx]], 0.0f);   // relu(act) before softmax
      amax = fmaxf(amax, a[cidx]);
    }
    float w[4], wsum = 0.0f;
    for (int cidx = 0; cidx < 4; ++cidx) { w[cidx] = expf(a[cidx] - amax); wsum += w[cidx]; }
    float inv = 1.0f / wsum;
    float xo = 0.0f;
    for (int cidx = 0; cidx < 4; ++cidx) xo += (w[cidx] * inv) * XT[ch[cidx] * fo + o];
    x_out[p * fo + o] = xo;
    if (o == 0) {
      float pvx = 0.0f, pvy = 0.0f;
      for (int cidx = 0; cidx < 4; ++cidx) {
        float2 pc = pose_in[ch[cidx]];
        pvx += (w[cidx] * inv) * pc.x;
        pvy += (w[cidx] * inv) * pc.y;
      }
      float nrm = sqrtf(pvx * pvx + pvy * pvy);
      float px = pvx / (nrm + 1e-6f), py = pvy / (nrm + 1e-6f);
      float ag = 0.0f;
      for (int cidx = 0; cidx < 4; ++cidx) {
        float2 pc = pose_in[ch[cidx]];
        ag += (w[cidx] * inv) * (pc.x * px + pc.y * py);
      }
      pose_out[p] = make_float2(px, py);
      act_out[p] = 1.0f / (1.0f + expf(-ag));
    }
  }
}

// ---------------------------------------------------------------------------
// log_softmax over 10 classes + act copy, packed into d_out (tuple order).
// ---------------------------------------------------------------------------
__global__ void final_kernel(const float* __restrict__ logits, const float* __restrict__ act,
                             float* __restrict__ out, int Bn) {
  int i = blockIdx.x * blockDim.x + threadIdx.x;
  if (i >= Bn) return;
  const float* l = logits + i * 10;
  float m = l[0];
  for (int j = 1; j < 10; ++j) m = fmaxf(m, l[j]);
  float sum = 0.0f;
  for (int j = 0; j < 10; ++j) sum += expf(l[j] - m);
  float lse = logf(sum);
  for (int j = 0; j < 10; ++j) out[i * 10 + j] = l[j] - m - lse;
  out[Bn * 10 + i] = act[i];
}

// ---------------------------------------------------------------------------
extern "C" void kernel_launch(void* const* d_in, const int* in_sizes, int n_in,
                              void* d_out, int out_size, void* d_ws, size_t ws_size,
                              hipStream_t stream) {
  const float* img = (const float*)d_in[0];
  const float* Wconv[5];
  const float* Wpool[5];
  for (int i = 0; i < 5; ++i) {
    Wconv[i] = (const float*)d_in[1 + i];
    Wpool[i] = (const float*)d_in[6 + i];
  }

  float* ws = (float*)d_ws;
  // static arena (float offsets); total ~19.24M floats (~77 MB)
  float* poseA = ws + 0;         // 131072*2
  float* poseB = ws + 262144;    // 131072*2
  float* actA  = ws + 524288;    // 131072
  float* actB  = ws + 655360;    // 131072
  float* xA    = ws + 786432;    // 524288  (pool outputs / next-level x)
  float* xB    = ws + 1310720;   // 524288
  float* xconv = ws + 1835008;   // 2097152 (max: 131072*16)
  float* XT    = ws + 3932160;   // 2097152
  float* Bperm = ws + 6029312;   // 102400  (max: 25*64*64)
  float* Ybuf  = ws + 6131712;   // 13107200 (max: 32768*400, level 2)

  const int Bn = 128;
  const int Hs[5]   = {32, 16, 8, 4, 2};
  const int cis[5]  = {1, 16, 32, 32, 64};
  const int cos_[5] = {16, 16, 32, 32, 64};
  const int fos[5]  = {16, 32, 32, 64, 10};

  float* poseIn = poseA; float* poseOut = poseB;
  float* actIn = actA;   float* actOut = actB;
  float* xIn = xA;       float* xOut = xB;

  {
    int total = Bn * 1024;
    grad_kernel<<<(total + 255) / 256, 256, 0, stream>>>(img, (float2*)poseIn, actIn, total);
  }

  for (int lvl = 0; lvl < 5; ++lvl) {
    int H = Hs[lvl], N = Bn * H * H;          // N % 64 == 0 for all levels
    int ci = cis[lvl], co = cos_[lvl], fo = fos[lvl];

    if (lvl == 0) {
      int total = N * co;
      gconv_l1<<<(total + 255) / 256, 256, 0, stream>>>(img, (const float2*)poseIn,
                                                        Wconv[0], xconv, total);
    } else {
      int pn = 25 * ci * co;
      permW<<<(pn + 255) / 256, 256, 0, stream>>>(Wconv[lvl], Bperm, ci, co, pn);
      int NC = 25 * co;
      dim3 gg((NC + 15) / 16, N / 64);        // Kd = ci in {16,32,64}
      wmma_gemm_f32<<<gg, 128, 0, stream>>>(xIn, Bperm, Ybuf, N, ci, NC);
      int total = N * co;
      gconv_agg<<<(total + 255) / 256, 256, 0, stream>>>(Ybuf, (const float2*)poseIn,
                                                         xconv, H, co, total);
    }

    {
      dim3 gg((fo + 15) / 16, N / 64);        // Kd = co in {16,32,64}
      wmma_gemm_f32<<<gg, 128, 0, stream>>>(xconv, Wpool[lvl], XT, N, co, fo);
    }

    {
      int Np = N / 4;
      int total = Np * fo;
      pool_kernel<<<(total + 255) / 256, 256, 0, stream>>>(XT, (const float2*)poseIn, actIn,
                                                           xOut, (float2*)poseOut, actOut,
                                                           H, fo, total);
    }

    float* t;
    t = poseIn; poseIn = poseOut; poseOut = t;
    t = actIn;  actIn = actOut;   actOut = t;
    t = xIn;    xIn = xOut;       xOut = t;
  }

  final_kernel<<<1, 128, 0, stream>>>(xIn, actIn, (float*)d_out, Bn);
}